// LiftSplatShoot_29832842838031
// MI455X (gfx1250) — compile-verified
//
#include <hip/hip_runtime.h>

// ---------------- problem constants (from reference) ----------------
#define NX0 150
#define NX1 200
#define NBATCH 4
#define NBINS (NX0 * NX1 * NBATCH)   // 120000 (nx2 == 1, only iz==0 kept)
#define CFEAT 64

typedef __attribute__((ext_vector_type(2))) float v2f;
typedef __attribute__((ext_vector_type(8))) float v8f;

// ---------------- k0: zero bin counters ----------------
__global__ void k_zero(int* __restrict__ counts, int n) {
    int i = blockIdx.x * blockDim.x + threadIdx.x;
    if (i < n) counts[i] = 0;
}

// ---------------- k1: voxelize + histogram (1 int atomic / point) ----------------
__global__ void k_bin(const float* __restrict__ geom, int* __restrict__ vox,
                      int* __restrict__ counts, int nprime, int per_batch) {
    int p = blockIdx.x * blockDim.x + threadIdx.x;
    if (p >= nprime) return;
    const float* g = geom + (size_t)p * 3;
    float gx = g[0], gy = g[1], gz = g[2];
    // lo = bx - dx/2, computed in f32 exactly as the reference does
    const float lo0 = 0.3f    - 0.6f  * 0.5f;   // 0.0
    const float lo1 = -14.925f - 0.15f * 0.5f;  // -15.0
    const float lo2 = 0.0f    - 20.0f * 0.5f;   // -10.0
    // (int) cast truncates toward zero == astype(int32)
    int ix = (int)((gx - lo0) / 0.6f);
    int iy = (int)((gy - lo1) / 0.15f);
    int iz = (int)((gz - lo2) / 20.0f);
    bool kept = (ix >= 0) & (ix < NX0) & (iy >= 0) & (iy < NX1) & (iz >= 0) & (iz < 1);
    int b = p / per_batch;
    int bin = ix * (NX1 * NBATCH) + iy * NBATCH + iz * NBATCH + b;
    vox[p] = kept ? bin : -1;
    if (kept) atomicAdd(&counts[bin], 1);
}

// ---------------- k2a: per-block exclusive scan ----------------
__global__ void k_scan_block(const int* __restrict__ counts, int* __restrict__ offsets,
                             int* __restrict__ bsums, int n) {
    __shared__ int s[256];
    int i = blockIdx.x * 256 + threadIdx.x;
    int v = (i < n) ? counts[i] : 0;
    s[threadIdx.x] = v;
    __syncthreads();
    for (int d = 1; d < 256; d <<= 1) {
        int t = (threadIdx.x >= (unsigned)d) ? s[threadIdx.x - d] : 0;
        __syncthreads();
        s[threadIdx.x] += t;
        __syncthreads();
    }
    if (i < n) offsets[i] = s[threadIdx.x] - v;     // exclusive
    if (threadIdx.x == 255) bsums[blockIdx.x] = s[255];
}

// ---------------- k2b: scan the block sums (single block) ----------------
__global__ void k_scan_sums(int* __restrict__ bsums, int nb) {
    __shared__ int s[512];
    int v = (threadIdx.x < (unsigned)nb) ? bsums[threadIdx.x] : 0;
    s[threadIdx.x] = v;
    __syncthreads();
    for (int d = 1; d < 512; d <<= 1) {
        int t = (threadIdx.x >= (unsigned)d) ? s[threadIdx.x - d] : 0;
        __syncthreads();
        s[threadIdx.x] += t;
        __syncthreads();
    }
    if (threadIdx.x < (unsigned)nb) bsums[threadIdx.x] = s[threadIdx.x] - v;  // exclusive
}

// ---------------- k2c: add block offsets, re-zero counters for placement ----------------
__global__ void k_scan_add(int* __restrict__ offsets, int* __restrict__ counts,
                           const int* __restrict__ bsums, int n) {
    int i = blockIdx.x * 256 + threadIdx.x;
    if (i < n) { offsets[i] += bsums[blockIdx.x]; counts[i] = 0; }
}

// ---------------- k3: build per-bin point lists (1 int atomic / point) ----------------
__global__ void k_place(const int* __restrict__ vox, const int* __restrict__ offsets,
                        int* __restrict__ counts, int* __restrict__ plist, int nprime) {
    int p = blockIdx.x * blockDim.x + threadIdx.x;
    if (p >= nprime) return;
    int v = vox[p];
    if (v >= 0) {
        int pos = offsets[v] + atomicAdd(&counts[v], 1);
        plist[pos] = p;
    }
}

// ---------------- k4: gather-reduce, one wave32 per voxel bin, WMMA reduction ----------
// D = ones(16x4) * B(4x16) + C  ==> every row of D = sum over the 4 packed point-rows.
// 4 accumulators (16 features each) cover C=64. Result invariant to K-slot striping.
__global__ void __launch_bounds__(256) k_gather(const float* __restrict__ x,
                                                const int* __restrict__ offsets,
                                                const int* __restrict__ counts,
                                                const int* __restrict__ plist,
                                                float* __restrict__ out, int nbins) {
    int wid = blockIdx.x * (blockDim.x >> 5) + (threadIdx.x >> 5);  // uniform per wave
    if (wid >= nbins) return;

    // remap so consecutive waves -> consecutive iy (write coalescing in L2)
    int b  = wid / (NX0 * NX1);
    int r  = wid - b * (NX0 * NX1);
    int ix = r / NX1;
    int iy = r - ix * NX1;
    int bin = ix * (NX1 * NBATCH) + iy * NBATCH + b;

    int off = __builtin_amdgcn_readfirstlane(offsets[bin]);
    int cnt = __builtin_amdgcn_readfirstlane(counts[bin]);

    int lane = threadIdx.x & 31;
    int col  = lane & 15;
    bool hi  = lane >= 16;

    v2f ones; ones.x = 1.0f; ones.y = 1.0f;
    v8f a0 = {}, a1 = {}, a2 = {}, a3 = {};

    for (int j = 0; j < cnt; j += 4) {
        int p0 = (j + 0 < cnt) ? plist[off + j + 0] : -1;
        int p1 = (j + 1 < cnt) ? plist[off + j + 1] : -1;
        int p2 = (j + 2 < cnt) ? plist[off + j + 2] : -1;
        int p3 = (j + 3 < cnt) ? plist[off + j + 3] : -1;
        if (j + 4 < cnt) __builtin_prefetch(plist + off + j + 4, 0, 3);  // global_prefetch_b8

        int pa = hi ? p2 : p0;
        int pb = hi ? p3 : p1;
        const float* ra = x + (size_t)(pa > 0 ? pa : 0) * CFEAT;
        const float* rb = x + (size_t)(pb > 0 ? pb : 0) * CFEAT;
        float ma = (pa >= 0) ? 1.0f : 0.0f;
        float mb = (pb >= 0) ? 1.0f : 0.0f;

        v2f bm0, bm1, bm2, bm3;
        bm0.x = ma * ra[ 0 + col]; bm0.y = mb * rb[ 0 + col];
        bm1.x = ma * ra[16 + col]; bm1.y = mb * rb[16 + col];
        bm2.x = ma * ra[32 + col]; bm2.y = mb * rb[32 + col];
        bm3.x = ma * ra[48 + col]; bm3.y = mb * rb[48 + col];

        a0 = __builtin_amdgcn_wmma_f32_16x16x4_f32(false, ones, false, bm0, (short)0, a0, false, false);
        a1 = __builtin_amdgcn_wmma_f32_16x16x4_f32(false, ones, false, bm1, (short)0, a1, false, false);
        a2 = __builtin_amdgcn_wmma_f32_16x16x4_f32(false, ones, false, bm2, (short)0, a2, false, false);
        a3 = __builtin_amdgcn_wmma_f32_16x16x4_f32(false, ones, false, bm3, (short)0, a3, false, false);
    }

    // C/D layout: VGPR0, lanes 0..15 hold M=0, N=lane. All rows equal (A all-ones).
    if (lane < 16) {
        // out[b][c][ix][iy], c = g*16 + lane
        size_t xy = (size_t)ix * NX1 + iy;
        size_t plane = (size_t)NX0 * NX1;
        size_t base = ((size_t)b * CFEAT + lane) * plane + xy;
        out[base + (size_t)( 0) * plane] = a0[0];
        out[base + (size_t)(16) * plane] = a1[0];
        out[base + (size_t)(32) * plane] = a2[0];
        out[base + (size_t)(48) * plane] = a3[0];
    }
}

// ---------------- host side ----------------
extern "C" void kernel_launch(void* const* d_in, const int* in_sizes, int n_in,
                              void* d_out, int out_size, void* d_ws, size_t ws_size,
                              hipStream_t stream) {
    const float* geom = (const float*)d_in[0];   // [B,N,D,H,W,3]
    const float* x    = (const float*)d_in[1];   // [B,N,D,H,W,C]
    float* out        = (float*)d_out;           // [B, C, X, Y]
    int nprime = in_sizes[1] / CFEAT;
    int per_batch = nprime / NBATCH;

    // workspace layout (ints): counts | offsets | bsums | vox | plist
    int* ws      = (int*)d_ws;
    int* counts  = ws;
    int* offsets = counts + 120256;
    int* bsums   = offsets + 120256;
    int* vox     = bsums + 512;
    int* plist   = vox + nprime;

    const int NB = (NBINS + 255) / 256;   // 469 scan blocks (<= 512)

    k_zero<<<(NBINS + 255) / 256, 256, 0, stream>>>(counts, NBINS);
    k_bin<<<(nprime + 255) / 256, 256, 0, stream>>>(geom, vox, counts, nprime, per_batch);
    k_scan_block<<<NB, 256, 0, stream>>>(counts, offsets, bsums, NBINS);
    k_scan_sums<<<1, 512, 0, stream>>>(bsums, NB);
    k_scan_add<<<NB, 256, 0, stream>>>(offsets, counts, bsums, NBINS);
    k_place<<<(nprime + 255) / 256, 256, 0, stream>>>(vox, offsets, counts, plist, nprime);

    const int wavesPerBlock = 8;  // 256 threads = 8 wave32
    k_gather<<<(NBINS + wavesPerBlock - 1) / wavesPerBlock, 32 * wavesPerBlock, 0, stream>>>(
        x, offsets, counts, plist, out, NBINS);
}